// GCN_54348516163920
// MI455X (gfx1250) — compile-verified
//
#include <hip/hip_runtime.h>

typedef __attribute__((ext_vector_type(2))) float v2f;
typedef __attribute__((ext_vector_type(8))) float v8f;

#define N_NODES 100000
#define NFEAT   256
#define NHID    64
#define NCLASS  41

__global__ __launch_bounds__(256) void fill_zero_k(float* __restrict__ p, int n) {
    int i = blockIdx.x * blockDim.x + threadIdx.x;
    if (i < n) p[i] = 0.0f;
}

// ---------------- GEMM1: h1[N,64] = x[N,256] @ W1[256,64] (fp32 WMMA) -------------
__global__ __launch_bounds__(256) void gemm1_wmma_k(const float* __restrict__ x,
                                                    const float* __restrict__ W1,
                                                    float* __restrict__ h1) {
    const int lane = threadIdx.x & 31;
    const int wave = threadIdx.x >> 5;
    const int r0 = (blockIdx.x * 8 + wave) * 16;      // wave-uniform
    if (r0 >= N_NODES) return;                        // uniform exit: EXEC stays full

    const int m     = lane & 15;                      // M (for A) / N (for B/C)
    const int khalf = (lane >> 4) * 2;                // K sub-offset: 0 or 2

    int arow = r0 + m;
    if (arow >= N_NODES) arow = N_NODES - 1;          // clamped load, store guarded
    const float* __restrict__ xrow = x + (long)arow * NFEAT;

    v8f acc0 = {}, acc1 = {}, acc2 = {}, acc3 = {};
    for (int k0 = 0; k0 < NFEAT; k0 += 4) {
        v2f a;
        a.x = xrow[k0 + khalf];
        a.y = xrow[k0 + khalf + 1];
        const float* __restrict__ wlo = W1 + (k0 + khalf) * NHID + m;
        const float* __restrict__ whi = W1 + (k0 + khalf + 1) * NHID + m;
        v2f b0; b0.x = wlo[0];  b0.y = whi[0];
        v2f b1; b1.x = wlo[16]; b1.y = whi[16];
        v2f b2; b2.x = wlo[32]; b2.y = whi[32];
        v2f b3; b3.x = wlo[48]; b3.y = whi[48];
        acc0 = __builtin_amdgcn_wmma_f32_16x16x4_f32(false, a, false, b0, (short)0, acc0, false, false);
        acc1 = __builtin_amdgcn_wmma_f32_16x16x4_f32(false, a, false, b1, (short)0, acc1, false, false);
        acc2 = __builtin_amdgcn_wmma_f32_16x16x4_f32(false, a, false, b2, (short)0, acc2, false, false);
        acc3 = __builtin_amdgcn_wmma_f32_16x16x4_f32(false, a, false, b3, (short)0, acc3, false, false);
    }

    const int col     = lane & 15;
    const int rowHalf = (lane >> 4) * 8;
    for (int j = 0; j < 8; ++j) {
        int r = r0 + rowHalf + j;
        if (r < N_NODES) {
            float* __restrict__ o = h1 + (long)r * NHID + col;
            o[0]  = acc0[j];
            o[16] = acc1[j];
            o[32] = acc2[j];
            o[48] = acc3[j];
        }
    }
}

// ------- GEMM2: h2[N,41] = relu(agg1[N,64] + b1) @ W2[64,41] (fp32 WMMA) ---------
__global__ __launch_bounds__(256) void gemm2_wmma_k(const float* __restrict__ agg1,
                                                    const float* __restrict__ b1,
                                                    const float* __restrict__ W2,
                                                    float* __restrict__ h2) {
    const int lane = threadIdx.x & 31;
    const int wave = threadIdx.x >> 5;
    const int r0 = (blockIdx.x * 8 + wave) * 16;
    if (r0 >= N_NODES) return;

    const int m     = lane & 15;
    const int khalf = (lane >> 4) * 2;

    int arow = r0 + m;
    if (arow >= N_NODES) arow = N_NODES - 1;
    const float* __restrict__ hrow = agg1 + (long)arow * NHID;

    v8f acc0 = {}, acc1 = {}, acc2 = {};
    for (int k0 = 0; k0 < NHID; k0 += 4) {
        int ka = k0 + khalf;
        v2f a;
        a.x = fmaxf(hrow[ka]     + b1[ka],     0.0f);
        a.y = fmaxf(hrow[ka + 1] + b1[ka + 1], 0.0f);

        v2f b0, b1v, b2v;
        // N tiles 0..2 cover cols 0..47; zero-pad cols >= 41
        {
            int n = m;                  // tile 0: always < 41
            b0.x = W2[ka * NCLASS + n];
            b0.y = W2[(ka + 1) * NCLASS + n];
        }
        {
            int n = 16 + m;             // tile 1: always < 41
            b1v.x = W2[ka * NCLASS + n];
            b1v.y = W2[(ka + 1) * NCLASS + n];
        }
        {
            int n = 32 + m;             // tile 2: pad n >= 41 with 0
            int nc = (n < NCLASS) ? n : (NCLASS - 1);
            float lo = W2[ka * NCLASS + nc];
            float hi = W2[(ka + 1) * NCLASS + nc];
            if (n >= NCLASS) { lo = 0.0f; hi = 0.0f; }
            b2v.x = lo; b2v.y = hi;
        }
        acc0 = __builtin_amdgcn_wmma_f32_16x16x4_f32(false, a, false, b0,  (short)0, acc0, false, false);
        acc1 = __builtin_amdgcn_wmma_f32_16x16x4_f32(false, a, false, b1v, (short)0, acc1, false, false);
        acc2 = __builtin_amdgcn_wmma_f32_16x16x4_f32(false, a, false, b2v, (short)0, acc2, false, false);
    }

    const int col     = lane & 15;
    const int rowHalf = (lane >> 4) * 8;
    for (int j = 0; j < 8; ++j) {
        int r = r0 + rowHalf + j;
        if (r < N_NODES) {
            float* __restrict__ o = h2 + (long)r * NCLASS;
            o[col]      = acc0[j];
            o[col + 16] = acc1[j];
            int c2 = col + 32;
            if (c2 < NCLASS) o[c2] = acc2[j];
        }
    }
}

// ------------- SpMM: agg[dst[e], c] += w[e] * h[src[e], c], COO scatter -----------
template <int D>
__global__ __launch_bounds__(256) void spmm_k(const int* __restrict__ src,
                                              const int* __restrict__ dst,
                                              const float* __restrict__ w,
                                              const float* __restrict__ h,
                                              float* __restrict__ agg,
                                              int nEdges) {
    long idx = (long)blockIdx.x * blockDim.x + threadIdx.x;
    long total = (long)nEdges * D;
    if (idx >= total) return;
    int e = (int)(idx / D);
    int c = (int)(idx - (long)e * D);
    float v = w[e] * h[(long)src[e] * D + c];
    atomicAdd(&agg[(long)dst[e] * D + c], v);
}

// --------------------- out = log_softmax(agg2 + b2, axis=1) -----------------------
__global__ __launch_bounds__(256) void logsoftmax_k(const float* __restrict__ agg2,
                                                    const float* __restrict__ b2,
                                                    float* __restrict__ out) {
    int i = blockIdx.x * blockDim.x + threadIdx.x;
    if (i >= N_NODES) return;
    const float* __restrict__ z = agg2 + (long)i * NCLASS;
    float zb[NCLASS];
    float mx = -3.0e38f;
    for (int c = 0; c < NCLASS; ++c) {
        zb[c] = z[c] + b2[c];
        mx = fmaxf(mx, zb[c]);
    }
    float s = 0.0f;
    for (int c = 0; c < NCLASS; ++c) s += expf(zb[c] - mx);
    float lse = mx + logf(s);
    float* __restrict__ o = out + (long)i * NCLASS;
    for (int c = 0; c < NCLASS; ++c) o[c] = zb[c] - lse;
}

extern "C" void kernel_launch(void* const* d_in, const int* in_sizes, int n_in,
                              void* d_out, int out_size, void* d_ws, size_t ws_size,
                              hipStream_t stream) {
    const float* x    = (const float*)d_in[0];
    const int*   esrc = (const int*)d_in[1];
    const int*   edst = (const int*)d_in[2];
    const float* ew   = (const float*)d_in[3];
    const float* W1   = (const float*)d_in[4];
    const float* b1   = (const float*)d_in[5];
    const float* W2   = (const float*)d_in[6];
    const float* b2   = (const float*)d_in[7];
    float* out = (float*)d_out;
    const int E = in_sizes[1];

    float* h1   = (float*)d_ws;                         // N*64
    float* agg1 = h1   + (size_t)N_NODES * NHID;        // N*64
    float* h2   = agg1 + (size_t)N_NODES * NHID;        // N*41
    float* agg2 = h2   + (size_t)N_NODES * NCLASS;      // N*41

    const int n1 = N_NODES * NHID;
    const int n2 = N_NODES * NCLASS;
    fill_zero_k<<<(n1 + 255) / 256, 256, 0, stream>>>(agg1, n1);
    fill_zero_k<<<(n2 + 255) / 256, 256, 0, stream>>>(agg2, n2);

    const int gemmBlocks = (N_NODES + 127) / 128;       // 8 waves/block, 16 rows/wave
    gemm1_wmma_k<<<gemmBlocks, 256, 0, stream>>>(x, W1, h1);

    long w1 = (long)E * NHID;
    spmm_k<NHID><<<(unsigned)((w1 + 255) / 256), 256, 0, stream>>>(esrc, edst, ew, h1, agg1, E);

    gemm2_wmma_k<<<gemmBlocks, 256, 0, stream>>>(agg1, b1, W2, h2);

    long w2 = (long)E * NCLASS;
    spmm_k<NCLASS><<<(unsigned)((w2 + 255) / 256), 256, 0, stream>>>(esrc, edst, ew, h2, agg2, E);

    logsoftmax_k<<<(N_NODES + 255) / 256, 256, 0, stream>>>(agg2, b2, out);
}